// Attention_31172872634650
// MI455X (gfx1250) — compile-verified
//
#include <hip/hip_runtime.h>
#include <math.h>

typedef __attribute__((ext_vector_type(2))) float v2f;
typedef __attribute__((ext_vector_type(8))) float v8f;

#define B_      16
#define HQ_     32
#define HKV_    8
#define G_      4
#define D_      128
#define NSPLIT  8
#define PART_STRIDE (8 + G_*D_)   /* 4 m + 4 l + 4x128 acc (floats) */

#if defined(__has_builtin)
#if __has_builtin(__builtin_amdgcn_global_load_async_to_lds_b128)
#define HAVE_ASYNC_LDS 1
#endif
#endif

#ifdef HAVE_ASYNC_LDS
typedef int v4i __attribute__((vector_size(16)));
typedef __attribute__((address_space(1))) v4i Gv4i;   // global int4
typedef __attribute__((address_space(3))) v4i Lv4i;   // LDS int4

__device__ __forceinline__ void async_row_copy(const float* src, float* dstLds, int lane) {
    // one instruction: 32 lanes x 16B = one 512B row, global -> LDS, ASYNCcnt
    __builtin_amdgcn_global_load_async_to_lds_b128(
        (Gv4i*)(uintptr_t)(src + lane * 4),
        (Lv4i*)(uintptr_t)(dstLds + lane * 4), 0, 0);
}

__device__ __forceinline__ void wait_async_le32() {
#if __has_builtin(__builtin_amdgcn_s_wait_asynccnt)
    __builtin_amdgcn_s_wait_asynccnt(32);
#else
    asm volatile("s_wait_asynccnt 32" ::: "memory");
#endif
}
__device__ __forceinline__ void wait_async_0() {
#if __has_builtin(__builtin_amdgcn_s_wait_asynccnt)
    __builtin_amdgcn_s_wait_asynccnt(0);
#else
    asm volatile("s_wait_asynccnt 0" ::: "memory");
#endif
}
#endif // HAVE_ASYNC_LDS

// One wave (32 lanes) per (batch, kv_head, split). Streams 16-token K/V tiles
// through double-buffered LDS (async global->LDS copies), online-softmax flash
// attention with V_WMMA_F32_16X16X4_F32 for Q.K^T and P.V, writes one
// (m, l, acc) partial per split; a second kernel merges the splits.
__launch_bounds__(32)
__global__ void attn_partial_kernel(const float* __restrict__ xq,
                                    const float* __restrict__ xk,
                                    const float* __restrict__ xv,
                                    const float* __restrict__ kv_buffer,
                                    const int*   __restrict__ start_index,
                                    const int*   __restrict__ b_seq_len,
                                    const int*   __restrict__ max_seq_p,
                                    float*       __restrict__ part)
{
    __shared__ float ldsQ[16 * D_];        // 16 q rows (4 valid, rest 0), pre-scaled
    __shared__ float ldsK[2][16 * D_];     // double-buffered K tile: 16 seq x 128
    __shared__ float ldsV[2][16 * D_];     // double-buffered V tile
    __shared__ float ldsP[16 * 16];        // P^T tile: 16 seq x 16 q

    const int lane  = threadIdx.x;        // 0..31
    const int split = blockIdx.x;         // 0..NSPLIT-1
    const int h     = blockIdx.y;         // kv head 0..7
    const int b     = blockIdx.z;         // batch 0..15

    const int  maxseq = max_seq_p[0];
    const int  chunk  = (maxseq + NSPLIT - 1) / NSPLIT;   // 512 for 4096
    const int  limit  = b_seq_len[b];                     // valid length incl. new token
    const int  begin  = split * chunk;
    const int  endp   = (begin + chunk < limit) ? (begin + chunk) : limit;
    const long start  = (long)start_index[b];

    const float scale = 0.08838834764831845f;  // 1/sqrt(128)

    const int n   = lane & 15;          // B/C/D column (q index; 0..3 valid)
    const int hi  = lane >> 4;          // wave half
    const int off = hi * 2;             // K-slot offset per ISA A/B layout

    const size_t kvb_row = (size_t)(2 * HKV_) * D_;          // 2048 floats per cache row
    const float* krepl = xk + ((size_t)b * HKV_ + h) * D_;   // new-token K row
    const float* vrepl = xv + ((size_t)b * HKV_ + h) * D_;   // new-token V row

    // ---- stage Q (scaled) into LDS; rows 4..15 zero-padded -----------------
    {
        const float* qbase = xq + ((size_t)b * HQ_ + (size_t)h * G_) * D_;
        #pragma unroll
        for (int r = 0; r < 16; ++r) {
            float4 v = make_float4(0.f, 0.f, 0.f, 0.f);
            if (r < G_) {
                float4 q4 = *(const float4*)(qbase + (size_t)r * D_ + lane * 4);
                v = make_float4(q4.x * scale, q4.y * scale, q4.z * scale, q4.w * scale);
            }
            *(float4*)(&ldsQ[r * D_ + lane * 4]) = v;
        }
    }

    // ---- online softmax state ---------------------------------------------
    float m_run = -1e30f;
    float l_run = 0.f;
    v8f acc[8];                        // out^T accum: 8 d-chunks of 16 x (16 q)
    #pragma unroll
    for (int i = 0; i < 8; ++i)
        #pragma unroll
        for (int j = 0; j < 8; ++j) acc[i][j] = 0.f;

    const int nt = (endp > begin) ? ((endp - begin + 15) >> 4) : 0;

#ifdef HAVE_ASYNC_LDS
    // prologue: kick off tile 0 into buffer 0 (32 async row copies = 16 KB)
    if (nt > 0) {
        #pragma unroll
        for (int i = 0; i < 16; ++i) {
            const int s = begin + i;
            const float *krow, *vrow;
            if (s == limit - 1) { krow = krepl; vrow = vrepl; }
            else {
                const float* row = kv_buffer + (size_t)(start + s) * kvb_row;
                krow = row + (size_t)h * D_;
                vrow = row + (size_t)(HKV_ + h) * D_;
            }
            async_row_copy(krow, &ldsK[0][i * D_], lane);
            async_row_copy(vrow, &ldsV[0][i * D_], lane);
        }
    }
#endif

    for (int t = 0; t < nt; ++t) {
        const int s0  = begin + t * 16;
        const int cur = t & 1;

#ifdef HAVE_ASYNC_LDS
        // issue next tile into the other buffer, then wait for current tile
        if (t + 1 < nt) {
            const int s0n = s0 + 16;
            #pragma unroll
            for (int i = 0; i < 16; ++i) {
                const int s = s0n + i;
                const float *krow, *vrow;
                if (s == limit - 1) { krow = krepl; vrow = vrepl; }
                else {
                    const float* row = kv_buffer + (size_t)(start + s) * kvb_row;
                    krow = row + (size_t)h * D_;
                    vrow = row + (size_t)(HKV_ + h) * D_;
                }
                async_row_copy(krow, &ldsK[1 - cur][i * D_], lane);
                async_row_copy(vrow, &ldsV[1 - cur][i * D_], lane);
            }
            wait_async_le32();   // async loads retire in order: current tile landed
        } else {
            wait_async_0();
        }
#else
        // synchronous fallback: coalesced load -> LDS via VGPRs
        #pragma unroll
        for (int i = 0; i < 16; ++i) {
            const int s = s0 + i;
            const float *krow, *vrow;
            if (s == limit - 1) { krow = krepl; vrow = vrepl; }
            else {
                const float* row = kv_buffer + (size_t)(start + s) * kvb_row;
                krow = row + (size_t)h * D_;
                vrow = row + (size_t)(HKV_ + h) * D_;
            }
            *(float4*)(&ldsK[cur][i * D_ + lane * 4]) = *(const float4*)(krow + lane * 4);
            *(float4*)(&ldsV[cur][i * D_ + lane * 4]) = *(const float4*)(vrow + lane * 4);
        }
        __syncthreads();
#endif

        // ---- scores^T = K_tile (16xD) . Q^T (Dx16): 32 WMMA over d ---------
        v8f c;
        #pragma unroll
        for (int j = 0; j < 8; ++j) c[j] = 0.f;
        {
            const float* arow = &ldsK[cur][n * D_ + off];  // A: M = lane&15 (seq)
            const float* brow = &ldsQ[n * D_ + off];       // B: N = lane&15 (q)
            #pragma unroll
            for (int d0 = 0; d0 < D_; d0 += 4) {
                v2f a  = *(const v2f*)(arow + d0);
                v2f bb = *(const v2f*)(brow + d0);
                c = __builtin_amdgcn_wmma_f32_16x16x4_f32(false, a, false, bb,
                                                          (short)0, c, false, false);
            }
        }

        // ---- mask, tile max, exp, tile sum (columns split across halves) ---
        float tmax = -1e30f;
        #pragma unroll
        for (int r = 0; r < 8; ++r) {
            const int s = s0 + r + 8 * hi;                 // C row M = r + 8*hi
            float sc = (s < endp) ? c[r] : -1e30f;
            c[r] = sc;
            tmax = fmaxf(tmax, sc);
        }
        tmax = fmaxf(tmax, __shfl_xor(tmax, 16, 32));
        const float m_new = fmaxf(m_run, tmax);
        const float alpha = __expf(m_run - m_new);

        float lsum = 0.f;
        #pragma unroll
        for (int r = 0; r < 8; ++r) {
            float p = __expf(c[r] - m_new);
            lsum += p;
            ldsP[(r + 8 * hi) * 16 + n] = p;               // P^T to LDS for B operand
        }
        lsum += __shfl_xor(lsum, 16, 32);
        l_run = l_run * alpha + lsum;
        m_run = m_new;

        #pragma unroll
        for (int dc = 0; dc < 8; ++dc)
            #pragma unroll
            for (int r = 0; r < 8; ++r) acc[dc][r] *= alpha;

        // ---- out^T += V^T (Dx16) . P^T (16x16): 8 d-chunks x 4 K-steps -----
        #pragma unroll
        for (int dc = 0; dc < 8; ++dc) {
            #pragma unroll
            for (int k4 = 0; k4 < 4; ++k4) {
                const int sl = k4 * 4 + off;
                v2f a;                                      // A: M = d (lane&15)
                a.x = ldsV[cur][sl * D_ + dc * 16 + n];
                a.y = ldsV[cur][(sl + 1) * D_ + dc * 16 + n];
                v2f bb;                                     // B: N = q (lane&15)
                bb.x = ldsP[sl * 16 + n];
                bb.y = ldsP[(sl + 1) * 16 + n];
                acc[dc] = __builtin_amdgcn_wmma_f32_16x16x4_f32(false, a, false, bb,
                                                                (short)0, acc[dc],
                                                                false, false);
            }
        }
    }

    // ---- write split partial: m[4], l[4], acc[4][128] ----------------------
    float* pb = part + (size_t)((b * HKV_ + h) * NSPLIT + split) * PART_STRIDE;
    if (hi == 0 && n < G_) {
        pb[n]      = m_run;
        pb[G_ + n] = l_run;
    }
    if (n < G_) {
        #pragma unroll
        for (int dc = 0; dc < 8; ++dc) {
            float4 lo = make_float4(acc[dc][0], acc[dc][1], acc[dc][2], acc[dc][3]);
            float4 hv = make_float4(acc[dc][4], acc[dc][5], acc[dc][6], acc[dc][7]);
            float* dst = pb + 8 + n * D_ + dc * 16 + 8 * hi;   // d = dc*16 + 8*hi + r
            *(float4*)(dst)     = lo;
            *(float4*)(dst + 4) = hv;
        }
    }
}

// Merge NSPLIT partials per (b, kv_head): one block per (b,h), thread = d.
__launch_bounds__(128)
__global__ void attn_reduce_kernel(const float* __restrict__ part,
                                   float* __restrict__ out)
{
    const int bh = blockIdx.x;           // b*HKV + h
    const int d  = threadIdx.x;          // 0..127
    const int b  = bh / HKV_;
    const int h  = bh % HKV_;
    const float* pb0 = part + (size_t)bh * NSPLIT * PART_STRIDE;

    #pragma unroll
    for (int n = 0; n < G_; ++n) {
        float m_tot = -1e30f;
        #pragma unroll
        for (int i = 0; i < NSPLIT; ++i)
            m_tot = fmaxf(m_tot, pb0[i * PART_STRIDE + n]);
        float l_tot = 0.f, val = 0.f;
        #pragma unroll
        for (int i = 0; i < NSPLIT; ++i) {
            const float* p = pb0 + i * PART_STRIDE;
            const float w = __expf(p[n] - m_tot);      // 0 for empty splits
            l_tot += w * p[G_ + n];
            val   += w * p[8 + n * D_ + d];
        }
        out[((size_t)b * HQ_ + h * G_ + n) * D_ + d] = val / l_tot;
    }
}

extern "C" void kernel_launch(void* const* d_in, const int* in_sizes, int n_in,
                              void* d_out, int out_size, void* d_ws, size_t ws_size,
                              hipStream_t stream) {
    const float* xq        = (const float*)d_in[0];
    const float* xk        = (const float*)d_in[1];
    const float* xv        = (const float*)d_in[2];
    const float* kv_buffer = (const float*)d_in[3];
    // d_in[4] = cur_select_index (== start + seq_len - 1; derived in-kernel)
    const int*   start_idx = (const int*)d_in[5];
    const int*   b_seq_len = (const int*)d_in[6];
    const int*   max_seq   = (const int*)d_in[7];

    float* part = (float*)d_ws;   // 16*8*8 * 520 floats ~= 2.1 MB

    dim3 g1(NSPLIT, HKV_, B_);
    attn_partial_kernel<<<g1, 32, 0, stream>>>(xq, xk, xv, kv_buffer,
                                               start_idx, b_seq_len, max_seq, part);
    attn_reduce_kernel<<<B_ * HKV_, 128, 0, stream>>>(part, (float*)d_out);
}